// TopKSparsemaxWrapper_79250736546110
// MI455X (gfx1250) — compile-verified
//
#include <hip/hip_runtime.h>
#include <math.h>

// Problem constants (match reference)
#define B_ROWS 2048
#define D_DIM  1024
#define L_DIM  4096
#define K_TOP  10
#define NSUB   1024   // 2^K_TOP

typedef __attribute__((ext_vector_type(2))) float v2f;
typedef __attribute__((ext_vector_type(8))) float v8f;

#ifndef __has_builtin
#define __has_builtin(x) 0
#endif
#if __has_builtin(__builtin_amdgcn_wmma_f32_16x16x4_f32)
#define HAVE_WMMA_F32 1
#else
#define HAVE_WMMA_F32 0
#endif

// ---------------------------------------------------------------------------
// Kernel 0: zero the entropy accumulator slot (graph-replay safe)
// ---------------------------------------------------------------------------
__global__ void zero_ent_kernel(float* ent) { *ent = 0.0f; }

// ---------------------------------------------------------------------------
// Kernel 1: scores = x @ W + b  via V_WMMA_F32_16X16X4_F32 (exact fp32)
//   block = 256 threads = 8 waves; wave tile 32x64 (2 A-tiles x 4 B-tiles,
//   8 accumulators) so each B operand is reused by 2 WMMAs -> 10 loads per
//   8 WMMA instead of 9 per 4. Workgroup tile 64 rows x 256 cols;
//   grid (L/256, B/64) = (16, 32).
// ---------------------------------------------------------------------------
#if HAVE_WMMA_F32
__global__ __launch_bounds__(256) void gemm_scores_wmma(
    const float* __restrict__ x, const float* __restrict__ W,
    const float* __restrict__ bias, float* __restrict__ scores)
{
  const int lane   = threadIdx.x & 31;
  const int wave   = threadIdx.x >> 5;   // 0..7
  const int rowgrp = wave >> 2;          // 0..1
  const int colgrp = wave & 3;           // 0..3
  const int hi     = lane >> 4;          // 0/1  (K-half for A/B operands)
  const int m      = lane & 15;          // row (A) / col (B,C)

  const int row0 = blockIdx.y * 64  + rowgrp * 32;   // wave covers 32 rows
  const int col0 = blockIdx.x * 256 + colgrp * 64;   // wave covers 64 cols

  v8f acc00 = {}, acc01 = {}, acc02 = {}, acc03 = {};  // rows row0..row0+15
  v8f acc10 = {}, acc11 = {}, acc12 = {}, acc13 = {};  // rows row0+16..row0+31

  const float* xrow0 = x + (size_t)(row0 + m) * D_DIM;
  const float* xrow1 = xrow0 + (size_t)16 * D_DIM;

#pragma unroll 4
  for (int k0 = 0; k0 < D_DIM; k0 += 4) {
    // A (16x4 fp32): lanes 0-15 hold K=0,1; lanes 16-31 hold K=2,3
    v2f a0 = *(const v2f*)(xrow0 + k0 + 2 * hi);
    v2f a1 = *(const v2f*)(xrow1 + k0 + 2 * hi);

    // B (4x16 fp32): lanes 0-15 rows K=0,1 (N=lane); lanes 16-31 rows K=2,3
    const float* w0 = W + (size_t)(k0 + 2 * hi) * L_DIM + col0 + m;
    const float* w1 = w0 + L_DIM;
    if ((k0 & 31) == 0)  // keep the W stream warm in L2 (global_prefetch_b8)
      __builtin_prefetch(w0 + (size_t)32 * L_DIM, 0, 0);

    v2f b0 = { w0[0],  w1[0]  };
    v2f b1 = { w0[16], w1[16] };
    v2f b2 = { w0[32], w1[32] };
    v2f b3 = { w0[48], w1[48] };

    acc00 = __builtin_amdgcn_wmma_f32_16x16x4_f32(false, a0, false, b0, (short)0, acc00, false, false);
    acc10 = __builtin_amdgcn_wmma_f32_16x16x4_f32(false, a1, false, b0, (short)0, acc10, false, false);
    acc01 = __builtin_amdgcn_wmma_f32_16x16x4_f32(false, a0, false, b1, (short)0, acc01, false, false);
    acc11 = __builtin_amdgcn_wmma_f32_16x16x4_f32(false, a1, false, b1, (short)0, acc11, false, false);
    acc02 = __builtin_amdgcn_wmma_f32_16x16x4_f32(false, a0, false, b2, (short)0, acc02, false, false);
    acc12 = __builtin_amdgcn_wmma_f32_16x16x4_f32(false, a1, false, b2, (short)0, acc12, false, false);
    acc03 = __builtin_amdgcn_wmma_f32_16x16x4_f32(false, a0, false, b3, (short)0, acc03, false, false);
    acc13 = __builtin_amdgcn_wmma_f32_16x16x4_f32(false, a1, false, b3, (short)0, acc13, false, false);
  }

  const float bv0 = bias[col0 + m];
  const float bv1 = bias[col0 + 16 + m];
  const float bv2 = bias[col0 + 32 + m];
  const float bv3 = bias[col0 + 48 + m];

  // C/D layout: VGPR r -> M = r + 8*hi (lanes 0-15: M=r, lanes 16-31: M=r+8)
#pragma unroll
  for (int r = 0; r < 8; ++r) {
    const int row = row0 + 8 * hi + r;
    float* out0 = scores + (size_t)row * L_DIM + col0 + m;
    float* out1 = out0 + (size_t)16 * L_DIM;
    out0[0]  = acc00[r] + bv0;
    out0[16] = acc01[r] + bv1;
    out0[32] = acc02[r] + bv2;
    out0[48] = acc03[r] + bv3;
    out1[0]  = acc10[r] + bv0;
    out1[16] = acc11[r] + bv1;
    out1[32] = acc12[r] + bv2;
    out1[48] = acc13[r] + bv3;
  }
}
#else
// Fallback (should not be needed on gfx1250): one thread per output element
__global__ __launch_bounds__(256) void gemm_scores_naive(
    const float* __restrict__ x, const float* __restrict__ W,
    const float* __restrict__ bias, float* __restrict__ scores)
{
  const size_t idx = (size_t)blockIdx.x * blockDim.x + threadIdx.x;
  const int row = (int)(idx / L_DIM);
  const int col = (int)(idx % L_DIM);
  float acc = bias[col];
  for (int k = 0; k < D_DIM; ++k)
    acc = fmaf(x[(size_t)row * D_DIM + k], W[(size_t)k * L_DIM + col], acc);
  scores[idx] = acc;
}
#endif

// ---------------------------------------------------------------------------
// Kernel 2: per batch-row top-K / subsets / sparsemax / entropy / z stores
//   grid = B_ROWS blocks x 256 threads; LDS ~39 KB per workgroup
// ---------------------------------------------------------------------------
__global__ __launch_bounds__(256) void topk_sparsemax_kernel(
    const float* __restrict__ scores, float* __restrict__ z,
    float* __restrict__ distr, float* __restrict__ entropy)
{
  __shared__ float s_val[L_DIM];     // scores row
  __shared__ float s_abs[L_DIM];     // |scores| working copy (marked)
  __shared__ float s_sub[NSUB];      // subset sums (marked)
  __shared__ float s_rv[256];
  __shared__ int   s_ri[256];
  __shared__ float s_small[K_TOP];   // 10 smallest |s|, ascending
  __shared__ int   s_sidx[K_TOP];    // their indices
  __shared__ int   s_scode[K_TOP];   // 10 cheapest subset codes, ascending cost
  __shared__ float s_cost[K_TOP];
  __shared__ float s_P;

  const int tid  = threadIdx.x;
  const int brow = blockIdx.x;
  const float* srow = scores + (size_t)brow * L_DIM;

  // ---- load row into LDS, partial sum of positive part -------------------
  float psum = 0.0f;
#pragma unroll
  for (int q = 0; q < L_DIM / 256; ++q) {
    const int j = tid + 256 * q;
    const float v = srow[j];
    s_val[j] = v;
    s_abs[j] = fabsf(v);
    psum += (v > 0.0f) ? v : 0.0f;
  }
  s_rv[tid] = psum;
  __syncthreads();
  for (int off = 128; off > 0; off >>= 1) {
    if (tid < off) s_rv[tid] += s_rv[tid + off];
    __syncthreads();
  }
  if (tid == 0) s_P = s_rv[0];
  __syncthreads();

  // ---- 10 smallest |s| via repeated block argmin -------------------------
  for (int p = 0; p < K_TOP; ++p) {
    float best = 1e30f; int bidx = L_DIM;
#pragma unroll
    for (int q = 0; q < L_DIM / 256; ++q) {
      const int j = tid + 256 * q;
      const float v = s_abs[j];
      if (v < best || (v == best && j < bidx)) { best = v; bidx = j; }
    }
    s_rv[tid] = best; s_ri[tid] = bidx;
    __syncthreads();
    for (int off = 128; off > 0; off >>= 1) {
      if (tid < off) {
        const float v = s_rv[tid + off]; const int i2 = s_ri[tid + off];
        if (v < s_rv[tid] || (v == s_rv[tid] && i2 < s_ri[tid])) { s_rv[tid] = v; s_ri[tid] = i2; }
      }
      __syncthreads();
    }
    if (tid == 0) {
      s_small[p] = s_rv[0];
      s_sidx[p]  = s_ri[0];
      s_abs[s_ri[0]] = 1e30f;   // remove from further consideration
    }
    __syncthreads();
  }

  // ---- all 2^10 subset sums ----------------------------------------------
#pragma unroll
  for (int q = 0; q < NSUB / 256; ++q) {
    const int c = tid + 256 * q;
    float sum = 0.0f;
#pragma unroll
    for (int i = 0; i < K_TOP; ++i)
      if (c & (1 << i)) sum += s_small[i];
    s_sub[c] = sum;
  }
  __syncthreads();

  // ---- 10 cheapest subsets via repeated block argmin ---------------------
  for (int p = 0; p < K_TOP; ++p) {
    float best = 1e30f; int bidx = NSUB;
#pragma unroll
    for (int q = 0; q < NSUB / 256; ++q) {
      const int c = tid + 256 * q;
      const float v = s_sub[c];
      if (v < best || (v == best && c < bidx)) { best = v; bidx = c; }
    }
    s_rv[tid] = best; s_ri[tid] = bidx;
    __syncthreads();
    for (int off = 128; off > 0; off >>= 1) {
      if (tid < off) {
        const float v = s_rv[tid + off]; const int i2 = s_ri[tid + off];
        if (v < s_rv[tid] || (v == s_rv[tid] && i2 < s_ri[tid])) { s_rv[tid] = v; s_ri[tid] = i2; }
      }
      __syncthreads();
    }
    if (tid == 0) {
      s_cost[p]  = s_rv[0];
      s_scode[p] = s_ri[0];
      s_sub[s_ri[0]] = 1e30f;
    }
    __syncthreads();
  }

  // ---- sparsemax over sk[k] = P - cost[k]; entropy ------------------------
  if (tid == 0) {
    float sk[K_TOP], srt[K_TOP];
#pragma unroll
    for (int k = 0; k < K_TOP; ++k) { sk[k] = s_P - s_cost[k]; srt[k] = sk[k]; }
    // insertion sort, descending
    for (int i = 1; i < K_TOP; ++i) {
      const float v = srt[i]; int j = i - 1;
      while (j >= 0 && srt[j] < v) { srt[j + 1] = srt[j]; --j; }
      srt[j + 1] = v;
    }
    float cs = 0.0f, csAt = srt[0]; int support = 1;
    for (int r = 1; r <= K_TOP; ++r) {
      cs += srt[r - 1];
      if (1.0f + (float)r * srt[r - 1] > cs) { support = r; csAt = cs; }
    }
    const float tau = (csAt - 1.0f) / (float)support;
    float ent = 0.0f;
#pragma unroll
    for (int k = 0; k < K_TOP; ++k) {
      float d = sk[k] - tau;
      d = (d > 0.0f) ? d : 0.0f;
      if (d > 0.0f) ent -= d * logf(d);
      distr[brow * K_TOP + k] = d;
    }
    atomicAdd(entropy, ent * (1.0f / (float)B_ROWS));
  }

  // ---- write z: base bitvector x 10 rows, coalesced float4 stores ---------
#pragma unroll
  for (int k = 0; k < K_TOP; ++k) {
    float* zrow = z + ((size_t)brow * K_TOP + k) * L_DIM;
#pragma unroll
    for (int q = 0; q < L_DIM / (256 * 4); ++q) {
      const int j = (tid + 256 * q) * 4;
      float4 o;
      o.x = (s_val[j + 0] > 0.0f) ? 1.0f : 0.0f;
      o.y = (s_val[j + 1] > 0.0f) ? 1.0f : 0.0f;
      o.z = (s_val[j + 2] > 0.0f) ? 1.0f : 0.0f;
      o.w = (s_val[j + 3] > 0.0f) ? 1.0f : 0.0f;
      *(float4*)(zrow + j) = o;
    }
  }

  __threadfence_block();
  __syncthreads();

  // ---- apply <=100 flip fixups (k-th subset flips bit-set coords) ---------
  if (tid < K_TOP * K_TOP) {
    const int k = tid / K_TOP;
    const int i = tid % K_TOP;
    if (s_scode[k] & (1 << i)) {
      const int j = s_sidx[i];
      const float basev = (s_val[j] > 0.0f) ? 1.0f : 0.0f;
      z[((size_t)brow * K_TOP + k) * L_DIM + j] = 1.0f - basev;
    }
  }
}

// ---------------------------------------------------------------------------
// Launch
// ---------------------------------------------------------------------------
extern "C" void kernel_launch(void* const* d_in, const int* in_sizes, int n_in,
                              void* d_out, int out_size, void* d_ws, size_t ws_size,
                              hipStream_t stream) {
  const float* x    = (const float*)d_in[0];  // [B, D]
  const float* W    = (const float*)d_in[1];  // [D, L]
  const float* bias = (const float*)d_in[2];  // [L]

  float* out    = (float*)d_out;
  float* scores = (float*)d_ws;               // [B, L] fp32 scratch (32 MB)

  const size_t Z_ELEMS = (size_t)B_ROWS * K_TOP * L_DIM;
  float* z     = out;
  float* distr = out + Z_ELEMS;
  float* ent   = out + Z_ELEMS + (size_t)B_ROWS * K_TOP;

  zero_ent_kernel<<<1, 1, 0, stream>>>(ent);

#if HAVE_WMMA_F32
  dim3 g1(L_DIM / 256, B_ROWS / 64);
  gemm_scores_wmma<<<g1, 256, 0, stream>>>(x, W, bias, scores);
#else
  gemm_scores_naive<<<(B_ROWS * (size_t)L_DIM) / 256, 256, 0, stream>>>(x, W, bias, scores);
#endif

  topk_sparsemax_kernel<<<B_ROWS, 256, 0, stream>>>(scores, z, distr, ent);
}